// ExactLogPODE_20504173871579
// MI455X (gfx1250) — compile-verified
//
#include <hip/hip_runtime.h>
#include <math.h>

typedef float v2f __attribute__((ext_vector_type(2)));
typedef float v8f __attribute__((ext_vector_type(8)));

namespace {

constexpr int DD   = 16;   // ambient dim
constexpr int HH   = 128;  // hidden dim
constexpr int ROWS = 128;  // batch rows per block
constexpr int XS   = 20;   // sX row stride (dwords): [y(16) | s | 0 0 0] (K padded to 20)
constexpr int HS   = 18;   // per-wave tanh staging row stride (even -> 8B aligned v2f)
constexpr int AOS  = 17;   // sA row stride

struct Smem {
  float W1[20 * HH];      // rows 0..16 = W1 (17x128), rows 17..19 zero (K pad)
  float W2[HH * DD];      // 128x16 row-major
  float B1[HH];
  float B2[DD];
  float DW[HH];           // DW[k] = sum_{j<16} W1[j][k] * W2[k][j]  (for tr(Ja))
  float X[ROWS * XS];     // MLP input rows
  float Hs[8 * 16 * HS];  // per-wave h staging: D-layout -> LDS -> A-layout relay
  float A[ROWS * AOS];    // MLP output a (B x 16)
  float TP[ROWS * 16];    // per-(row,lane) divergence partials: tr(Ja) - m^T Ja m
};

// CDNA5 hardware transcendental tanh (V_TANH_F32) when available; this keeps
// the activation a single TRANS32 op that co-executes with the WMMA pipe,
// instead of OCML's branchy exp-based expansion (EXEC juggling between WMMAs).
__device__ __forceinline__ float fast_tanhf(float x) {
#if __has_builtin(__builtin_amdgcn_tanhf)
  return __builtin_amdgcn_tanhf(x);
#elif __has_builtin(__builtin_amdgcn_tanh_f32)
  return __builtin_amdgcn_tanh_f32(x);
#else
  return tanhf(x);
#endif
}

__device__ __forceinline__ v8f wmma4(v2f a, v2f b, v8f c) {
  // D = A(16x4 f32) * B(4x16 f32) + C(16x16 f32)
  return __builtin_amdgcn_wmma_f32_16x16x4_f32(false, a, false, b, (short)0, c,
                                               false, false);
}

// Full MLP eval for this wave's 16-row tile via V_WMMA_F32_16X16X4_F32.
// WITH_DIV additionally computes q = m @ W2^T and the divergence partials.
template <bool WITH_DIV>
__device__ __forceinline__ void mlp_wave_eval(Smem& sm, int wv, int half,
                                              int ln, float s) {
  const int row0 = wv * 16;
  float* hs = &sm.Hs[wv * 16 * HS];

  v8f oacc;  // layer-2 accumulator (16 rows x 16 out dims), C init = b2
  {
    const float bb2 = sm.B2[ln];
#pragma unroll
    for (int j = 0; j < 8; ++j) oacc[j] = bb2;
  }
  float tp[8];
#pragma unroll
  for (int j = 0; j < 8; ++j) tp[j] = 0.0f;

#pragma unroll
  for (int nt = 0; nt < 8; ++nt) {
    const int n0 = nt * 16;
    const int kcol = n0 + ln;

    // ---- layer 1: P = [y|s|0] @ W1 + b1   (M=16 rows, N=16 hidden, K=20) ----
    v8f p;
    {
      const float bb1 = sm.B1[kcol];
#pragma unroll
      for (int j = 0; j < 8; ++j) p[j] = bb1;
    }
#pragma unroll
    for (int kc = 0; kc < 5; ++kc) {
      const int k = kc * 4 + 2 * half;  // ISA A/B f32 layout: vgpr v <-> K = k0+2*half+v
      v2f a = *(const v2f*)&sm.X[(row0 + ln) * XS + k];
      v2f b;
      b[0] = sm.W1[k * HH + kcol];
      b[1] = sm.W1[(k + 1) * HH + kcol];
      p = wmma4(a, b, p);
    }
    v8f h;
#pragma unroll
    for (int j = 0; j < 8; ++j) h[j] = fast_tanhf(p[j]);
    // stage h (D-layout) into LDS for the layer-2 A operand (A-layout reload)
#pragma unroll
    for (int j = 0; j < 8; ++j) hs[(j + 8 * half) * HS + ln] = h[j];

    if (WITH_DIV) {
      // q[r][k] = W2[k,:] . m   (i.e. m @ W2^T), same D-layout as p
      v8f q;
#pragma unroll
      for (int j = 0; j < 8; ++j) q[j] = 0.0f;
#pragma unroll
      for (int jc = 0; jc < 4; ++jc) {
        const int kk = jc * 4 + 2 * half;
        v2f a = *(const v2f*)&sm.X[(row0 + ln) * XS + kk];
        v2f b = *(const v2f*)&sm.W2[kcol * DD + kk];  // W2^T fragment
        q = wmma4(a, b, q);
      }
      const float w1s = sm.W1[16 * HH + kcol];  // W1 time row
      const float dwk = sm.DW[kcol];
      const float b1k = sm.B1[kcol];
#pragma unroll
      for (int j = 0; j < 8; ++j) {
        const float hv  = h[j];
        const float omh = 1.0f - hv * hv;
        const float rr  = p[j] - b1k - s * w1s;  // m . W1[0:16, k]
        tp[j] += omh * (dwk - q[j] * rr);        // tr(Ja) - m^T Ja m partial
      }
    }

    // ---- layer 2 partial: a += h @ W2 for this n-tile's K chunk ----
#pragma unroll
    for (int kc = 0; kc < 4; ++kc) {
      const int kk = kc * 4 + 2 * half;
      v2f a = *(const v2f*)&hs[ln * HS + kk];
      v2f b;
      b[0] = sm.W2[(n0 + kk) * DD + ln];
      b[1] = sm.W2[(n0 + kk + 1) * DD + ln];
      oacc = wmma4(a, b, oacc);
    }
  }

#pragma unroll
  for (int j = 0; j < 8; ++j) sm.A[(row0 + j + 8 * half) * AOS + ln] = oacc[j];
  if (WITH_DIV) {
#pragma unroll
    for (int j = 0; j < 8; ++j) sm.TP[(row0 + j + 8 * half) * 16 + ln] = tp[j];
  }
}

__global__ __launch_bounds__(256) void exact_logp_ode_kernel(
    const float* __restrict__ x, const float* __restrict__ t,
    const float* __restrict__ gW1, const float* __restrict__ gb1,
    const float* __restrict__ gW2, const float* __restrict__ gb2,
    const int* __restrict__ gsteps, float* __restrict__ out) {
  __shared__ Smem sm;
  const int tid  = (int)threadIdx.x;
  const int wv   = tid >> 5;
  const int lane = tid & 31;
  const int half = lane >> 4;
  const int ln   = lane & 15;

  // ---- one-time: weights -> LDS ----
  for (int i = tid; i < 20 * HH; i += 256) sm.W1[i] = (i < 17 * HH) ? gW1[i] : 0.0f;
  for (int i = tid; i < HH * DD; i += 256) sm.W2[i] = gW2[i];
  if (tid < HH) sm.B1[tid] = gb1[tid];
  if (tid < DD) sm.B2[tid] = gb2[tid];
  __syncthreads();
  if (tid < HH) {
    float acc = 0.0f;
    for (int j = 0; j < DD; ++j) acc += sm.W1[j * HH + tid] * sm.W2[tid * DD + j];
    sm.DW[tid] = acc;
  }

  const int grow = (int)blockIdx.x * ROWS + tid;
  float y[16];
  if (tid < ROWS) {
    for (int j = 0; j < 16; ++j) y[j] = x[grow * 16 + j];
    sm.X[tid * XS + 17] = 0.0f;
    sm.X[tid * XS + 18] = 0.0f;
    sm.X[tid * XS + 19] = 0.0f;
  }
  float logp = 0.0f;
  const int nsteps = gsteps[0];
  const float ds = t[0] / (float)nsteps;
  __syncthreads();

  for (int it = 0; it < nsteps; ++it) {
    const float s = ds * (float)it;
    float nrm = 1.0f;

    // ---- divergence at m = y/||y|| (exact reduction of jacfwd trace):
    //   div = (0.5/||y||) * ( tr(Ja) - m^T Ja m - (d-1)*(a . m) )
    if (tid < ROWS) {
      float ss = 0.0f;
      for (int j = 0; j < 16; ++j) ss += y[j] * y[j];
      nrm = sqrtf(ss);
      const float inv = 1.0f / nrm;
      for (int j = 0; j < 16; ++j) sm.X[tid * XS + j] = y[j] * inv;
      sm.X[tid * XS + 16] = s;
    }
    __syncthreads();
    mlp_wave_eval<true>(sm, wv, half, ln, s);
    __syncthreads();
    if (tid < ROWS) {
      float t12 = 0.0f;
      for (int l = 0; l < 16; ++l) t12 += sm.TP[tid * 16 + l];
      float adot = 0.0f;
      const float inv = 1.0f / nrm;
      for (int j = 0; j < 16; ++j) adot += sm.A[tid * AOS + j] * (y[j] * inv);
      const float div = (0.5f / nrm) * (t12 - (float)(DD - 1) * adot);
      logp -= div * ds;
    }

    // ---- RK4 on drift(y) = P_y( -0.5*(d-1)*y - 0.5*a(y,s) ) ----
    float yt[16], acc[16];
    if (tid < ROWS) {
      for (int j = 0; j < 16; ++j) { yt[j] = y[j]; acc[j] = 0.0f; }
    }
    const float soff[4] = {0.0f, 0.5f, 0.5f, 1.0f};
    const float wgt[4]  = {1.0f, 2.0f, 2.0f, 1.0f};
    for (int st = 0; st < 4; ++st) {
      if (tid < ROWS) {
        for (int j = 0; j < 16; ++j) sm.X[tid * XS + j] = yt[j];
        sm.X[tid * XS + 16] = s + soff[st] * ds;
      }
      __syncthreads();
      mlp_wave_eval<false>(sm, wv, half, ln, s);
      __syncthreads();
      if (tid < ROWS) {
        float k[16];
        float gy = 0.0f, yy = 0.0f;
        for (int j = 0; j < 16; ++j) {
          const float g = -0.5f * (float)(DD - 1) * yt[j] - 0.5f * sm.A[tid * AOS + j];
          k[j] = g;
          gy += g * yt[j];
          yy += yt[j] * yt[j];
        }
        const float c = gy / yy;  // proj2tangent
        for (int j = 0; j < 16; ++j) {
          k[j] -= c * yt[j];
          acc[j] += wgt[st] * k[j];
        }
        if (st < 3) {
          const float cf = (st == 2) ? 1.0f : 0.5f;
          for (int j = 0; j < 16; ++j) yt[j] = y[j] + cf * ds * k[j];
        }
      }
    }
    if (tid < ROWS) {
      float nn = 0.0f;
      for (int j = 0; j < 16; ++j) {
        y[j] += (ds / 6.0f) * acc[j];
        nn += y[j] * y[j];
      }
      const float inv = 1.0f / sqrtf(nn);  // proj2manifold
      for (int j = 0; j < 16; ++j) y[j] *= inv;
    }
    __syncthreads();
  }

  if (tid < ROWS) {
    // logp0 = -(log 2 + 8*log(pi) - lgamma(8)) for S^15
    const float logp0 = -1.3258249f;
    out[grow] = logp0 + logp;
  }
}

}  // namespace

extern "C" void kernel_launch(void* const* d_in, const int* in_sizes, int n_in,
                              void* d_out, int out_size, void* d_ws, size_t ws_size,
                              hipStream_t stream) {
  const float* x  = (const float*)d_in[0];
  const float* t  = (const float*)d_in[1];
  const float* W1 = (const float*)d_in[2];
  const float* b1 = (const float*)d_in[3];
  const float* W2 = (const float*)d_in[4];
  const float* b2 = (const float*)d_in[5];
  const int* steps = (const int*)d_in[6];
  float* out = (float*)d_out;

  const int B = in_sizes[0] / DD;  // 1024
  dim3 grid(B / ROWS), block(256);
  exact_logp_ode_kernel<<<grid, block, 0, stream>>>(x, t, W1, b1, W2, b2, steps, out);
  (void)d_ws; (void)ws_size; (void)n_in; (void)out_size;
}